// _RNN_34041910788197
// MI455X (gfx1250) — compile-verified
//
#include <hip/hip_runtime.h>
#include <hip/hip_bf16.h>
#include <math.h>

#define L_SEQ 524288
#define H 64

typedef __attribute__((ext_vector_type(2))) float v2f;
typedef __attribute__((ext_vector_type(8))) float v8f;

__device__ __forceinline__ float fast_tanh(float x) {
#if __has_builtin(__builtin_amdgcn_tanhf)
    return __builtin_amdgcn_tanhf(x);
#elif __has_builtin(__builtin_amdgcn_tanh_f32)
    return __builtin_amdgcn_tanh_f32(x);
#else
    return tanhf(x);
#endif
}

// ---------------------------------------------------------------------------
// Kernel 1: pre0[t][i] = x[t]·W_ih0[i] + (b_ih0[i] + b_hh0[i]).  K=3, pure BW.
// ---------------------------------------------------------------------------
__global__ __launch_bounds__(256) void rnn_pre0(
    const float* __restrict__ x, const float* __restrict__ Wih0,
    const float* __restrict__ bih0, const float* __restrict__ bhh0,
    float* __restrict__ pre0) {
    long gid = (long)blockIdx.x * blockDim.x + threadIdx.x;  // = t*64 + i
    int i = (int)(gid & 63);
    long t = gid >> 6;
    const float* xr = x + t * 3;
    const float* w = Wih0 + i * 3;
    float acc = bih0[i] + bhh0[i];
    acc += xr[0] * w[0] + xr[1] * w[1] + xr[2] * w[2];
    pre0[gid] = acc;
}

// ---------------------------------------------------------------------------
// Kernel 2: serial scan, layer 0.  One 64-thread block (2 waves).
// Lane i owns h[i]; W_hh row i lives in 16 float4 registers; h broadcast via
// LDS (16x ds_load_b128 per step).  pre[t+1] load is issued one step early so
// its L2 latency hides behind the FMA chain + barriers.
// ---------------------------------------------------------------------------
__global__ __launch_bounds__(64) void rnn_scan_store(
    const float* __restrict__ pre, const float* __restrict__ Whh,
    float* __restrict__ hseq) {
    __shared__ float sh[H];
    const int i = threadIdx.x;
    float4 w4[H / 4];
    const float4* wr = (const float4*)(Whh + i * H);
#pragma unroll
    for (int j = 0; j < H / 4; ++j) w4[j] = wr[j];
    sh[i] = 0.0f;
    __syncthreads();
    float pcur = pre[i];                       // pre[0][i]
    for (long t = 0; t < L_SEQ; ++t) {
        float pnext = 0.0f;
        if (t + 1 < L_SEQ) pnext = pre[(t + 1) * H + i];   // issued early
        float acc = pcur;
        const float4* s4 = (const float4*)sh;
#pragma unroll
        for (int j = 0; j < H / 4; ++j) {
            float4 hv = s4[j];
            acc += w4[j].x * hv.x + w4[j].y * hv.y + w4[j].z * hv.z + w4[j].w * hv.w;
        }
        float hn = fast_tanh(acc);
        hseq[t * H + i] = hn;          // feeds the parallel WMMA GEMM later
        __syncthreads();               // all lanes done reading old h
        sh[i] = hn;
        pcur = pnext;
        __syncthreads();               // new h visible to all lanes
    }
}

// ---------------------------------------------------------------------------
// Kernel 3: pre1 = h0 @ W_ih1^T + bias via v_wmma_f32_16x16x4_f32.
// Wave = one 16-row (timestep) block x 64 cols (4 tiles of 16), K=64 as 16 x K4.
// A layout (16x4 f32): lanes 0-15 -> {K0,K1}, lanes 16-31 -> {K2,K3}.
// B layout (4x16 f32): lanes->N, lanes 0-15 -> rows {K0,K1}, 16-31 -> {K2,K3}.
// D layout: VGPR r -> row r (lanes 0-15) / row r+8 (lanes 16-31), N = lane%16.
// ---------------------------------------------------------------------------
__global__ __launch_bounds__(256) void rnn_pre1_wmma(
    const float* __restrict__ h0, const float* __restrict__ Wih1,
    const float* __restrict__ bih1, const float* __restrict__ bhh1,
    float* __restrict__ pre1) {
    const int lane = threadIdx.x & 31;
    const int wave = threadIdx.x >> 5;
    const int l = lane & 15;
    const int half = lane >> 4;
    const long mblk = (long)blockIdx.x * 8 + wave;  // 16-timestep block
    const long row = mblk * 16 + l;

    // Preload the 16 A chunks (K = kc*4 .. kc*4+3) for this row block.
    v2f a[16];
#pragma unroll
    for (int kc = 0; kc < 16; ++kc) {
        a[kc] = *(const v2f*)(h0 + row * H + kc * 4 + 2 * half);
    }

#pragma unroll
    for (int cb = 0; cb < 4; ++cb) {
        const int n = cb * 16 + l;
        v8f c = {0.f, 0.f, 0.f, 0.f, 0.f, 0.f, 0.f, 0.f};
#pragma unroll
        for (int kc = 0; kc < 16; ++kc) {
            // B[k][n] = W_ih1[n][k]  (pre1[t][n] = sum_k h0[t][k]*W_ih1[n][k])
            v2f b = *(const v2f*)(Wih1 + n * H + kc * 4 + 2 * half);
            c = __builtin_amdgcn_wmma_f32_16x16x4_f32(
                /*neg_a=*/false, a[kc], /*neg_b=*/false, b,
                /*c_mod=*/(short)0, c, /*reuse_a=*/false, /*reuse_b=*/false);
        }
        const float bv = bih1[n] + bhh1[n];
#pragma unroll
        for (int r = 0; r < 8; ++r) {
            const long orow = mblk * 16 + r + 8 * half;
            pre1[orow * H + n] = c[r] + bv;
        }
    }
}

// ---------------------------------------------------------------------------
// Kernel 4: serial scan, layer 1, + final Linear(H,1) on the last hidden.
// ---------------------------------------------------------------------------
__global__ __launch_bounds__(64) void rnn_scan_final(
    const float* __restrict__ pre, const float* __restrict__ Whh,
    const float* __restrict__ Wout, const float* __restrict__ bout,
    float* __restrict__ out) {
    __shared__ float sh[H];
    __shared__ float red[H];
    const int i = threadIdx.x;
    float4 w4[H / 4];
    const float4* wr = (const float4*)(Whh + i * H);
#pragma unroll
    for (int j = 0; j < H / 4; ++j) w4[j] = wr[j];
    sh[i] = 0.0f;
    __syncthreads();
    float hn = 0.0f;
    float pcur = pre[i];                       // pre[0][i]
    for (long t = 0; t < L_SEQ; ++t) {
        float pnext = 0.0f;
        if (t + 1 < L_SEQ) pnext = pre[(t + 1) * H + i];   // issued early
        float acc = pcur;
        const float4* s4 = (const float4*)sh;
#pragma unroll
        for (int j = 0; j < H / 4; ++j) {
            float4 hv = s4[j];
            acc += w4[j].x * hv.x + w4[j].y * hv.y + w4[j].z * hv.z + w4[j].w * hv.w;
        }
        hn = fast_tanh(acc);
        __syncthreads();
        sh[i] = hn;
        pcur = pnext;
        __syncthreads();
    }
    red[i] = Wout[i] * hn;
    __syncthreads();
    if (i == 0) {
        float s = bout[0];
#pragma unroll
        for (int j = 0; j < H; ++j) s += red[j];
        out[0] = s;
    }
}

// ---------------------------------------------------------------------------
extern "C" void kernel_launch(void* const* d_in, const int* in_sizes, int n_in,
                              void* d_out, int out_size, void* d_ws, size_t ws_size,
                              hipStream_t stream) {
    const float* x    = (const float*)d_in[0];
    const float* Wih0 = (const float*)d_in[1];
    const float* Whh0 = (const float*)d_in[2];
    const float* bih0 = (const float*)d_in[3];
    const float* bhh0 = (const float*)d_in[4];
    const float* Wih1 = (const float*)d_in[5];
    const float* Whh1 = (const float*)d_in[6];
    const float* bih1 = (const float*)d_in[7];
    const float* bhh1 = (const float*)d_in[8];
    const float* Wout = (const float*)d_in[9];
    const float* bout = (const float*)d_in[10];

    float* buf_pre = (float*)d_ws;                            // pre0, reused as pre1
    float* buf_h0  = (float*)d_ws + (size_t)L_SEQ * H;        // layer-0 hidden sequence

    // 1) parallel input projection for layer 0
    rnn_pre0<<<(L_SEQ * H) / 256, 256, 0, stream>>>(x, Wih0, bih0, bhh0, buf_pre);
    // 2) serial layer-0 scan (writes full h0 sequence)
    rnn_scan_store<<<1, 64, 0, stream>>>(buf_pre, Whh0, buf_h0);
    // 3) parallel WMMA GEMM: pre1 = h0 @ W_ih1^T + bias (overwrites pre0 buffer)
    rnn_pre1_wmma<<<L_SEQ / (16 * 8), 256, 0, stream>>>(buf_h0, Wih1, bih1, bhh1, buf_pre);
    // 4) serial layer-1 scan + output head
    rnn_scan_final<<<1, 64, 0, stream>>>(buf_pre, Whh1, Wout, bout, (float*)d_out);
}